// Memory_88682484728626
// MI455X (gfx1250) — compile-verified
//
#include <hip/hip_runtime.h>
#include <hip/hip_bf16.h>
#include <math.h>

// ---------------------------------------------------------------------------
// DNC memory module forward for MI455X (gfx1250, wave32).
// B=64, M=8192, C=64, D=512, W=1, R=4.
// Bandwidth-bound (~400-540 MB HBM @ 23.3 TB/s ~= 20 us). Strategy:
//   K1: controller GEMMs (tiny) + activations + key norms
//   K2: per-batch-row allocation: bitonic sort + product scan in LDS, scatter
//   K3: fused stream: new_mem = mem*reset + write, row L2 norm, and
//       dot[b,r,m] = keys . new_mem  via V_WMMA_F32_16X16X4_F32 (K=C=64 as
//       16 k-steps of 4), B-fragments fed directly from just-computed new_mem
//   K4: softmax over M + read_words = read_w . new_mem, streaming new_mem
//       tiles through LDS with GLOBAL_LOAD_ASYNC_TO_LDS_B128 (double-buffered,
//       ASYNCcnt-tracked) to kill the 4x redundant per-head L1 reads.
// ---------------------------------------------------------------------------

constexpr int   B_  = 64;
constexpr int   M_  = 8192;
constexpr int   C_  = 64;
constexpr int   D_  = 512;
constexpr int   R_  = 4;
constexpr float EPS = 1e-6f;

typedef __attribute__((ext_vector_type(2))) float v2f;
typedef __attribute__((ext_vector_type(8))) float v8f;

// --------------------------- K1: controller ---------------------------------
// One block per batch row, 512 threads. Each of the 389 output neurons is a
// 512-long dot product against inputs[b] staged in LDS.
__global__ void ctrl_kernel(const float* __restrict__ inputs,
                            const float* __restrict__ Wwv, const float* __restrict__ bwv,
                            const float* __restrict__ Wev, const float* __restrict__ bev,
                            const float* __restrict__ Wwg, const float* __restrict__ bwg,
                            const float* __restrict__ Wrk, const float* __restrict__ brk,
                            const float* __restrict__ Wrs, const float* __restrict__ brs,
                            float* __restrict__ wv, float* __restrict__ ev,
                            float* __restrict__ wg, float* __restrict__ keys,
                            float* __restrict__ strengths, float* __restrict__ knorm) {
  __shared__ float x[D_];
  __shared__ float sk[R_ * C_];
  const int b = blockIdx.x, t = threadIdx.x;
  x[t] = inputs[b * D_ + t];
  __syncthreads();

  if (t < 389) {
    const float* wrow;
    float bias;
    if (t < 64)        { wrow = Wwv + t * D_;         bias = bwv[t]; }
    else if (t < 128)  { wrow = Wev + (t - 64) * D_;  bias = bev[t - 64]; }
    else if (t == 128) { wrow = Wwg;                  bias = bwg[0]; }
    else if (t < 385)  { wrow = Wrk + (t - 129) * D_; bias = brk[t - 129]; }
    else               { wrow = Wrs + (t - 385) * D_; bias = brs[t - 385]; }
    float acc = bias;
    #pragma unroll 8
    for (int k = 0; k < D_; ++k) acc += wrow[k] * x[k];

    if (t < 64)        wv[b * C_ + t] = acc;
    else if (t < 128)  ev[b * C_ + (t - 64)] = 1.f / (1.f + __expf(-acc));
    else if (t == 128) wg[b] = 1.f / (1.f + __expf(-acc));
    else if (t < 385)  { keys[b * R_ * C_ + (t - 129)] = acc; sk[t - 129] = acc; }
    else               strengths[b * R_ + (t - 385)] =
                           (acc > 20.f) ? acc : log1pf(__expf(acc));
  }
  __syncthreads();
  if (t < R_) {
    float s = 0.f;
    for (int c = 0; c < C_; ++c) { float v = sk[t * C_ + c]; s += v * v; }
    knorm[b * R_ + t] = sqrtf(s + EPS);
  }
}

// --------------------------- K2: allocation ---------------------------------
// One block per batch row, 1024 threads. LDS: nonusage[8192] + idx[8192] +
// two scan buffers = 128 KB (WGP LDS is 320 KB). Bitonic sort descending by
// nonusage (tie: lower index first, matching jax top_k stability), then
// Hillis-Steele product scan for the exclusive cumprod, then scatter through
// the sorted index (== take_along_axis with the inverse permutation).
__global__ void alloc_kernel(const float* __restrict__ usage,
                             const float* __restrict__ wg,
                             float* __restrict__ ww,
                             float* __restrict__ new_usage) {
  extern __shared__ float smem[];
  float* sval = smem;                 // [M_] sorted nonusage
  int*   sidx = (int*)(smem + M_);    // [M_]
  float* bufA = smem + 2 * M_;        // [M_]
  float* bufB = smem + 3 * M_;        // [M_]
  const int b = blockIdx.x, t = threadIdx.x, NT = blockDim.x;

  for (int j = t; j < M_; j += NT) {
    float u  = usage[b * M_ + j];
    float uc = EPS + (1.f - EPS) * u;
    sval[j] = 1.f - uc;
    sidx[j] = j;
  }
  __syncthreads();

  // bitonic sort
  for (int k = 2; k <= M_; k <<= 1) {
    for (int jj = k >> 1; jj > 0; jj >>= 1) {
      for (int i = t; i < M_; i += NT) {
        int p = i ^ jj;
        if (p > i) {
          bool dirDesc = ((i & k) == 0);
          float vi = sval[i], vp = sval[p];
          int   ii = sidx[i], ip = sidx[p];
          bool pBeforeI = (vp > vi) || (vp == vi && ip < ii);
          bool doswap = dirDesc ? pBeforeI : !pBeforeI;
          if (doswap) { sval[i] = vp; sval[p] = vi; sidx[i] = ip; sidx[p] = ii; }
        }
      }
      __syncthreads();
    }
  }

  // inclusive product scan of sorted usage (1 - sorted_nonusage)
  for (int j = t; j < M_; j += NT) bufA[j] = 1.f - sval[j];
  __syncthreads();
  float* src = bufA; float* dst = bufB;
  for (int off = 1; off < M_; off <<= 1) {
    for (int j = t; j < M_; j += NT) {
      float v = src[j];
      if (j >= off) v *= src[j - off];
      dst[j] = v;
    }
    __syncthreads();
    float* tmp = src; src = dst; dst = tmp;
  }
  // inclusive scan now in src; exclusive[j] = (j ? src[j-1] : 1)

  const float wgb = wg[b];
  for (int j = t; j < M_; j += NT) {
    float excl  = (j == 0) ? 1.f : src[j - 1];
    float alloc = sval[j] * excl;
    int   p     = sidx[j];
    float w     = wgb * alloc;            // write_w (W == 1)
    ww[b * M_ + p] = w;
    float u0 = usage[b * M_ + p];
    new_usage[b * M_ + p] = u0 + (1.f - u0) * w;  // prod(1-ww) == 1-ww for W=1
  }
}

// --------------------- K3: fused update + WMMA dot --------------------------
// One wave per 16 memory rows. lane&15 selects the row, lane>>4 selects which
// channel pair (c%4 in {0,1} vs {2,3}) per the f32 16x16x4 A/B VGPR striping:
//   A 16x4: VGPR0 lanes0-15 K=0 / lanes16-31 K=2; VGPR1 K=1 / K=3 (M=lane%16)
//   B 4x16: same K striping, N=lane%16
//   D 16x16: VGPR d, lanes0-15 -> M=d, N=lane (heads r=0..3 live in d0..d3)
// EXEC stays all-ones through the WMMA loop (masking via select, not branch).
__global__ void mem_update_dot_kernel(const float* __restrict__ memory,
                                      const float* __restrict__ ww,
                                      const float* __restrict__ erase,
                                      const float* __restrict__ wvec,
                                      const float* __restrict__ keys,
                                      float* __restrict__ new_mem,
                                      float* __restrict__ dotp,
                                      float* __restrict__ mnorm) {
  const int lane = threadIdx.x & 31;
  const int wave = threadIdx.x >> 5;           // 0..7
  const int b    = blockIdx.x >> 6;            // grid = B * 64 blocks
  const int blk  = blockIdx.x & 63;
  const int tile = blk * 8 + wave;             // 0..511
  const int m0   = tile * 16;
  const int row  = m0 + (lane & 15);
  const int half = lane >> 4;                  // 0: c%4 in {0,1}; 1: {2,3}

  const float w = ww[(size_t)b * M_ + row];
  const float* __restrict__ memrow = memory  + ((size_t)b * M_ + row) * C_;
  float*       __restrict__ nmrow  = new_mem + ((size_t)b * M_ + row) * C_;
  const float* __restrict__ er  = erase + b * C_;
  const float* __restrict__ wvb = wvec  + b * C_;

  const int   rA    = lane & 15;
  const float maskA = (rA < R_) ? 1.f : 0.f;
  const int   rc    = (rA < R_) ? rA : 0;
  const float* __restrict__ krow = keys + ((size_t)b * R_ + rc) * C_;

  v8f acc = {};
  float sq = 0.f;

  #pragma unroll
  for (int kk = 0; kk < 16; ++kk) {
    const int c0 = 4 * kk + 2 * half;
    // A fragment: keys (rows >= R zeroed by mask, not by EXEC)
    v2f a;
    a.x = maskA * krow[c0];
    a.y = maskA * krow[c0 + 1];
    // B fragment: freshly computed new_mem values
    float mv0 = memrow[c0], mv1 = memrow[c0 + 1];
    float n0 = mv0 * (1.f - er[c0] * w)     + w * wvb[c0];
    float n1 = mv1 * (1.f - er[c0 + 1] * w) + w * wvb[c0 + 1];
    nmrow[c0]     = n0;
    nmrow[c0 + 1] = n1;
    sq += n0 * n0 + n1 * n1;
    v2f bf; bf.x = n0; bf.y = n1;
    acc = __builtin_amdgcn_wmma_f32_16x16x4_f32(
        /*neg_a=*/false, a, /*neg_b=*/false, bf,
        /*c_mod=*/(short)0, acc, /*reuse_a=*/false, /*reuse_b=*/false);
  }

  // combine the two half-lane partial squared sums for the row norm
  sq += __shfl_xor(sq, 16, 32);
  if (half == 0) {
    mnorm[(size_t)b * M_ + row] = sqrtf(sq + EPS);
    // D: lanes 0-15, VGPR r -> dot for head r, memory row m0+lane
    const int n = lane;
    dotp[((size_t)b * R_ + 0) * M_ + m0 + n] = acc[0];
    dotp[((size_t)b * R_ + 1) * M_ + m0 + n] = acc[1];
    dotp[((size_t)b * R_ + 2) * M_ + m0 + n] = acc[2];
    dotp[((size_t)b * R_ + 3) * M_ + m0 + n] = acc[3];
  }
}

// --------------------------- K4: softmax + read -----------------------------
// One block per batch, 256 threads: thread = (r = t>>6, c/j = t&63).
// Two cheap passes over the [R,M] logits, then one stream of new_mem[b]
// (2 MB per block) staged through LDS with double-buffered async copies:
// GLOBAL_LOAD_ASYNC_TO_LDS_B128 (ASYNCcnt) -> each 16 KB tile is pulled from
// HBM/L2 exactly once, all 4 read-head groups consume it from LDS broadcast.
__global__ void read_kernel(const float* __restrict__ dotp,
                            const float* __restrict__ mnorm,
                            const float* __restrict__ knorm,
                            const float* __restrict__ strengths,
                            const float* __restrict__ new_mem,
                            float* __restrict__ out_read) {
  __shared__ float red[R_ * 64];
  __shared__ float rowmax[R_];
  __shared__ float rowsum[R_];
  __shared__ float pexp[R_][64];
  __shared__ float tile[2][64 * C_];   // 2 x 16 KB double buffer
  const int b = blockIdx.x, t = threadIdx.x;
  const int r = t >> 6, j = t & 63;

  const float kn = knorm[b * R_ + r];
  const float st = strengths[b * R_ + r];
  const float* __restrict__ drow = dotp + ((size_t)b * R_ + r) * M_;
  const float* __restrict__ mn   = mnorm + (size_t)b * M_;

  // pass A: row max of s = dot*strength / (knorm*mnorm + EPS)
  float mx = -3.4e38f;
  for (int m = j; m < M_; m += 64) {
    float s = drow[m] * st / (kn * mn[m] + EPS);
    mx = fmaxf(mx, s);
  }
  red[t] = mx;
  __syncthreads();
  for (int off = 32; off > 0; off >>= 1) {
    if (j < off) red[t] = fmaxf(red[t], red[t + off]);
    __syncthreads();
  }
  if (j == 0) rowmax[r] = red[t];
  __syncthreads();
  const float rmx = rowmax[r];

  // pass B: row sum of exp
  float sm = 0.f;
  for (int m = j; m < M_; m += 64) {
    float s = drow[m] * st / (kn * mn[m] + EPS);
    sm += __expf(s - rmx);
  }
  red[t] = sm;
  __syncthreads();
  for (int off = 32; off > 0; off >>= 1) {
    if (j < off) red[t] += red[t + off];
    __syncthreads();
  }
  if (j == 0) rowsum[r] = red[t];
  __syncthreads();
  const float inv = 1.f / rowsum[r];

  // pass C: read_words[r,c] = sum_m p(r,m) * new_mem[m,c]
  const int c = j;
  const float* __restrict__ nm = new_mem + (size_t)b * M_ * C_;

  // Issue 4 async B128 loads per thread: one 64x64 f32 tile = 16 KB.
  // vdst VGPR holds the wave-relative LDS byte address (low 32 bits of the
  // generic pointer per the LDS aperture rules); addr is a 64-bit VGPR pair.
  auto prefetch_tile = [&](int mt, int bufsel) {
    const char* gbase = (const char*)(nm + (size_t)mt * C_);
    unsigned    lbase = (unsigned)(size_t)(&tile[bufsel][0]);
    #pragma unroll
    for (int q = 0; q < 4; ++q) {
      unsigned    off = (unsigned)(t * 16 + q * 4096);
      unsigned    lda = lbase + off;
      const char* ga  = gbase + off;
      asm volatile("global_load_async_to_lds_b128 %0, %1, off"
                   :: "v"(lda), "v"(ga)
                   : "memory");
    }
  };

  prefetch_tile(0, 0);
  float acc = 0.f;
  for (int mt = 0, ti = 0; mt < M_; mt += 64, ++ti) {
    // overlap: issue next tile's async loads before consuming this one
    if (mt + 64 < M_) prefetch_tile(mt + 64, (ti + 1) & 1);
    float s = drow[mt + j] * st / (kn * mn[mt + j] + EPS);
    pexp[r][j] = __expf(s - rmx);
    // async loads complete in order: <=4 outstanding means this tile landed
    if (mt + 64 < M_)
      asm volatile("s_wait_asynccnt 0x4" ::: "memory");
    else
      asm volatile("s_wait_asynccnt 0x0" ::: "memory");
    __syncthreads();
    const float* __restrict__ tb = &tile[ti & 1][0];
    #pragma unroll 8
    for (int u = 0; u < 64; ++u)
      acc += pexp[r][u] * tb[u * 64 + c];
    __syncthreads();
  }
  out_read[((size_t)b * R_ + r) * C_ + c] = acc * inv;
}

// ----------------------------- launcher -------------------------------------
extern "C" void kernel_launch(void* const* d_in, const int* in_sizes, int n_in,
                              void* d_out, int out_size, void* d_ws, size_t ws_size,
                              hipStream_t stream) {
  (void)in_sizes; (void)n_in; (void)out_size; (void)ws_size;
  const float* inputs = (const float*)d_in[0];
  const float* memory = (const float*)d_in[1];
  const float* usage  = (const float*)d_in[2];
  const float* Wwv = (const float*)d_in[3];  const float* bwv = (const float*)d_in[4];
  const float* Wev = (const float*)d_in[5];  const float* bev = (const float*)d_in[6];
  const float* Wwg = (const float*)d_in[7];  const float* bwg = (const float*)d_in[8];
  const float* Wrk = (const float*)d_in[9];  const float* brk = (const float*)d_in[10];
  const float* Wrs = (const float*)d_in[11]; const float* brs = (const float*)d_in[12];

  float* out       = (float*)d_out;
  float* out_read  = out;                                  // [B,R,C]
  float* out_mem   = out + (size_t)B_ * R_ * C_;           // [B,M,C]
  float* out_usage = out_mem + (size_t)B_ * M_ * C_;       // [B,M]

  float* ws        = (float*)d_ws;
  float* wv        = ws;                         // B*C
  float* ev        = wv + B_ * C_;               // B*C
  float* wg        = ev + B_ * C_;               // B
  float* keys      = wg + B_;                    // B*R*C
  float* strengths = keys + B_ * R_ * C_;        // B*R
  float* knorm     = strengths + B_ * R_;        // B*R
  float* ww        = knorm + B_ * R_;            // B*M
  float* dotp      = ww + (size_t)B_ * M_;       // B*R*M
  float* mnorm     = dotp + (size_t)B_ * R_ * M_;// B*M

  ctrl_kernel<<<B_, D_, 0, stream>>>(inputs, Wwv, bwv, Wev, bev, Wwg, bwg,
                                     Wrk, brk, Wrs, brs,
                                     wv, ev, wg, keys, strengths, knorm);

  alloc_kernel<<<B_, 1024, 4 * M_ * (int)sizeof(float), stream>>>(
      usage, wg, ww, out_usage);

  mem_update_dot_kernel<<<B_ * 64, 256, 0, stream>>>(
      memory, ww, ev, wv, keys, out_mem, dotp, mnorm);

  read_kernel<<<B_, R_ * 64, 0, stream>>>(
      dotp, mnorm, knorm, strengths, out_mem, out_read);
}